// PolynomialG_8564164788356
// MI455X (gfx1250) — compile-verified
//
#include <hip/hip_runtime.h>
#include <hip/hip_bf16.h>

typedef __attribute__((ext_vector_type(2))) float v2f;
typedef __attribute__((ext_vector_type(8))) float v8f;

#define DECAY_THRESH 1e-4f

// ---------------------------------------------------------------------------
// Pass 1: serial nonlinear h-scan (one wave). Produces:
//   Z[t]  (= h carried INTO step t; Z[t+1] = h_new at step t; Z[0]=0)
//   reset bitmask (1 bit per step) in workspace
// Also zero-fills the t=0 rows of J and H.
// ---------------------------------------------------------------------------
__global__ void h_scan_kernel(const float* __restrict__ X,
                              const float* __restrict__ params,
                              float* out,
                              unsigned* __restrict__ flags,
                              int T) {
  const int lane = threadIdx.x;   // 32 threads = 1 wave (wave32)
  const int nsteps = T - 1;

  const float w1 = tanhf(params[0]);
  const float w2 = tanhf(params[1]);
  const float w3 = params[2];
  const float w4 = params[3];

  float* Z = out;
  if (lane == 0) Z[0] = 0.0f;
  if (lane < 4)  (out + (size_t)T)[lane] = 0.0f;          // J[0][:]
  if (lane < 16) (out + 5 * (size_t)T)[lane] = 0.0f;      // H[0][:]

  float h = 0.0f;
  float decay = 1.0f;

  for (int base = 0; base < nsteps; base += 32) {
    const int t = base + lane;
    const float x = (t < nsteps) ? X[t] : 0.0f;
    if (base + 512 < nsteps) __builtin_prefetch(&X[base + 512 + lane], 0, 0);
    // off-critical-path: a_t = w1*x + w3*x^2
    const float aL = fmaf(w3, x, w1) * x;

    float hkeep = 0.0f;
    bool  rbit  = false;
#pragma unroll
    for (int i = 0; i < 32; ++i) {
      const float ai   = __shfl(aL, i);
      const float dfdh = fmaf(2.0f * w4, h, w2);          // uses carry h
      const float hn   = fmaf(h, fmaf(w4, h, w2), ai);    // 2-deep chain
      decay *= fabsf(dfdh);
      const bool reset = decay < DECAY_THRESH;
      if (reset) decay = 1.0f;
      if (lane == i) { hkeep = hn; rbit = reset; }
      h = hn;
    }
    const unsigned mask = (unsigned)__ballot(rbit);
    if (t < nsteps) Z[t + 1] = hkeep;
    if (lane == 0) flags[base >> 5] = mask;
  }
}

// ---------------------------------------------------------------------------
// Pass 2: one wave per 16-step chunk. J/H propagation recast as triangular
// matmuls on the WMMA pipe:
//   J[16x4] = L[16x16] * Dw[16x4]  + K (.) Jcarry
//   H[16x16]= L[16x16] * S [16x16] + K (.) Hcarry
// each as 4 chained V_WMMA_F32_16X16X4_F32. Exact chunk carries via replay
// from the last reset before the chunk (avg ~6 steps).
// Dw and S are materialized row-wise in LDS (row = chunk step owned by lane
// idx, built purely from that lane's registers + one b128 Jold load), so all
// WMMA B-operand reads are real indexed ds_loads — no register-array extract
// chains.
// ---------------------------------------------------------------------------
__global__ void jh_wmma_kernel(const float* __restrict__ X,
                               const float* __restrict__ params,
                               float* out,
                               const unsigned* __restrict__ flags,
                               int T) {
  const int lane = threadIdx.x;          // 32 threads = 1 wave
  const int nsteps = T - 1;
  const int t0 = blockIdx.x * 16;

  const float w1      = tanhf(params[0]);
  const float dtanh1  = 1.0f - w1 * w1;
  const float d2tanh1 = -2.0f * w1 * dtanh1;
  const float w2      = tanhf(params[1]);
  const float dtanh2  = 1.0f - w2 * w2;
  const float d2tanh2 = -2.0f * w2 * dtanh2;
  const float w4      = params[3];

  const float* Z = out;
  float* Jout = out + (size_t)T;
  float* Hout = out + 5 * (size_t)T;

  const int  idx    = lane & 15;      // chunk-local step (A row) / B column
  const bool hiHalf = lane >= 16;
  const int  cl = idx;                // output column owned by this lane
  const int  ci = cl >> 2, cj = cl & 3;

  // per-lane constants for the replay loop (branchless g_i, Hww terms)
  const float ei = (ci == 1) ? dtanh2 : 0.0f;   // g_i = ei + fi*h
  const float fi = (ci == 3) ? 2.0f : 0.0f;
  const float ej = (cj == 1) ? dtanh2 : 0.0f;   // g_j = ej + fj*h
  const float fj = (cj == 3) ? 2.0f : 0.0f;
  const float hw0 = (cl == 0) ? d2tanh1 : 0.0f;
  const float hw5 = (cl == 5) ? d2tanh2 : 0.0f;

  // --- find last reset strictly before t0 (carry is zero right after it) ---
  int s0 = 0;
  if (t0 > 0) {
    int s = t0 - 1;
    int w = s >> 5;
    unsigned bits = flags[w] & (0xFFFFFFFFu >> (31 - (s & 31)));
    while (true) {
      if (bits) { s0 = (w << 5) + (31 - __clz((int)bits)) + 1; break; }
      if (w == 0) break;
      --w;
      bits = flags[w];
    }
  }

  // --- replay [s0, t0): J redundant across lanes, H distributed (comp cl) ---
  float Jv0 = 0.f, Jv1 = 0.f, Jv2 = 0.f, Jv3 = 0.f, Hc = 0.f;
  for (int s = s0; s < t0; ++s) {
    const float x  = X[s];
    const float hh = Z[s];                       // carry h into step s
    const float aS = fmaf(2.0f * w4, hh, w2);
    const float gi = fmaf(fi, hh, ei);
    const float gj = fmaf(fj, hh, ej);
    const float Ji = (ci == 0) ? Jv0 : (ci == 1) ? Jv1 : (ci == 2) ? Jv2 : Jv3;
    const float Jj = (cj == 0) ? Jv0 : (cj == 1) ? Jv1 : (cj == 2) ? Jv2 : Jv3;
    const float hw = hw0 * x + hw5 * hh;
    Hc  = fmaf(aS, Hc, hw + gi * Jj + Ji * gj);  // H uses pre-update J
    Jv0 = fmaf(aS, Jv0, x * dtanh1);
    Jv1 = fmaf(aS, Jv1, hh * dtanh2);
    Jv2 = fmaf(aS, Jv2, x * x);
    Jv3 = fmaf(aS, Jv3, hh * hh);
    if ((flags[s >> 5] >> (s & 31)) & 1u) { Jv0 = Jv1 = Jv2 = Jv3 = 0.f; Hc = 0.f; }
  }

  // --- LDS state ---
  // Jold[s][j] = J carried INTO chunk step s (row 0 = chunk carry).
  // DwL[s][c]  = df_dw matrix, cols 4..15 zero.   SL[s][c] = Hessian source.
  __shared__ float ms[16], asv[16], Karr[16], Jold[17][4];
  __shared__ float DwL[16][16], SL[16][16];

  // --- stage chunk-local per-step scalars; build Dw row from registers ---
  float xr, hr;                       // this lane's step (row idx) x and h
  {
    const int tg = t0 + idx;
    const bool valid = tg < nsteps;
    xr = valid ? X[tg] : 0.0f;
    hr = valid ? Z[tg] : 0.0f;
    const bool r = valid ? (((flags[tg >> 5] >> (tg & 31)) & 1u) != 0u) : true;
    if (!hiHalf) {
      ms[idx]  = (valid && !r) ? 1.0f : 0.0f;
      asv[idx] = fmaf(2.0f * w4, hr, w2);
      float4 d0 = make_float4(xr * dtanh1, hr * dtanh2, xr * xr, hr * hr);
      float4 zz = make_float4(0.f, 0.f, 0.f, 0.f);
      *(float4*)&DwL[idx][0]  = d0;
      *(float4*)&DwL[idx][4]  = zz;
      *(float4*)&DwL[idx][8]  = zz;
      *(float4*)&DwL[idx][12] = zz;
    }
    if (lane < 4)
      Jold[0][lane] = (lane == 0) ? Jv0 : (lane == 1) ? Jv1
                    : (lane == 2) ? Jv2 : Jv3;
  }
  __syncthreads();

  // --- build row `idx` of the triangular transfer matrix L, and K_t ---
  // L[t][s] = m_s * prod_{u=s+1..t}(m_u a_u);  L[t][s] = m_s a_{s+1} L[t][s+1]
  float Lrow[16];
  {
    const float m15 = ms[15];
    float val = (idx == 15) ? m15 : 0.0f;
    Lrow[15] = val;
#pragma unroll
    for (int s = 14; s >= 0; --s) {
      const float m_s = ms[s];
      const float a_n = asv[s + 1];
      const float cand = val * (m_s * a_n);
      val = (s == idx) ? m_s : ((s < idx) ? cand : 0.0f);
      Lrow[s] = val;
    }
  }
  float K = 1.0f;
#pragma unroll
  for (int u = 0; u < 16; ++u) {
    const float p = ms[u] * asv[u];
    K *= (u <= idx) ? p : 1.0f;
  }
  if (!hiHalf) Karr[idx] = K;

  const int koff = (lane >> 4) << 1;  // 0 for lanes 0-15, 2 for lanes 16-31

  // --- J = L * Dw : 4 chained WMMA f32 16x16x4 ---
  // A frag (16x4 slab of L): lanes0-15 -> {K=0,1}, lanes16-31 -> {K=2,3}
  // B frag: plain indexed LDS loads (no register-array extraction possible)
  v8f cJ = {};
#pragma unroll
  for (int k = 0; k < 4; ++k) {
    v2f A;
    A.x = hiHalf ? Lrow[4 * k + 2] : Lrow[4 * k + 0];
    A.y = hiHalf ? Lrow[4 * k + 3] : Lrow[4 * k + 1];
    const int sa = 4 * k + koff;
    v2f B;
    B.x = DwL[sa][cl];
    B.y = DwL[sa + 1][cl];
    cJ = __builtin_amdgcn_wmma_f32_16x16x4_f32(false, A, false, B,
                                               (short)0, cJ, false, false);
  }
  __syncthreads();   // Karr / Jold[0] writes visible before use

  // add carry term, publish J chunk to LDS (Jold[t+1] = J_out[t])
  const float Jc = (cl == 0) ? Jv0 : (cl == 1) ? Jv1
                 : (cl == 2) ? Jv2 : (cl == 3) ? Jv3 : 0.0f;
#pragma unroll
  for (int r8 = 0; r8 < 8; ++r8) {
    const int tr = r8 + (hiHalf ? 8 : 0);
    cJ[r8] += Karr[tr] * Jc;
  }
  if (cl < 4) {
#pragma unroll
    for (int r8 = 0; r8 < 8; ++r8) {
      const int tr = r8 + (hiHalf ? 8 : 0);
      Jold[tr + 1][cl] = cJ[r8];
    }
  }
  __syncthreads();

  // --- build S row `idx` from registers + one b128 Jold load ---
  // S[s][4i+j] = Hww(i,j) + g_i*Jold[s][j] + Jold[s][i]*g_j
  if (!hiHalf) {
    const float4 Jr = *(const float4*)&Jold[idx][0];
    const float Jv[4] = {Jr.x, Jr.y, Jr.z, Jr.w};
    const float g[4]  = {0.0f, dtanh2, 0.0f, 2.0f * hr};
#pragma unroll
    for (int i4 = 0; i4 < 4; ++i4) {
      float4 row;
      float* rp = (float*)&row;
#pragma unroll
      for (int j4 = 0; j4 < 4; ++j4) {
        float v = g[i4] * Jv[j4] + Jv[i4] * g[j4];
        if (i4 == 0 && j4 == 0) v += xr * d2tanh1;
        if (i4 == 1 && j4 == 1) v += hr * d2tanh2;
        rp[j4] = v;
      }
      *(float4*)&SL[idx][4 * i4] = row;
    }
  }
  __syncthreads();

  // --- H = L * S : 4 chained WMMA f32 16x16x4 ---
  v8f cH = {};
#pragma unroll
  for (int k = 0; k < 4; ++k) {
    v2f A;
    A.x = hiHalf ? Lrow[4 * k + 2] : Lrow[4 * k + 0];
    A.y = hiHalf ? Lrow[4 * k + 3] : Lrow[4 * k + 1];
    const int sa = 4 * k + koff;
    v2f B;
    B.x = SL[sa][cl];
    B.y = SL[sa + 1][cl];
    cH = __builtin_amdgcn_wmma_f32_16x16x4_f32(false, A, false, B,
                                               (short)0, cH, false, false);
  }
#pragma unroll
  for (int r8 = 0; r8 < 8; ++r8) {
    const int tr = r8 + (hiHalf ? 8 : 0);
    cH[r8] += Karr[tr] * Hc;
  }

  // --- stores: C layout VGPR r -> row r (lo half) / r+8 (hi half), col=cl ---
#pragma unroll
  for (int r8 = 0; r8 < 8; ++r8) {
    const int tr = r8 + (hiHalf ? 8 : 0);
    const int tg = t0 + tr;
    if (tg < nsteps) {
      if (cl < 4) Jout[(size_t)(tg + 1) * 4 + cl] = cJ[r8];
      Hout[(size_t)(tg + 1) * 16 + cl] = cH[r8];
    }
  }
}

// ---------------------------------------------------------------------------
extern "C" void kernel_launch(void* const* d_in, const int* in_sizes, int n_in,
                              void* d_out, int out_size, void* d_ws, size_t ws_size,
                              hipStream_t stream) {
  const float* X      = (const float*)d_in[0];
  const float* params = (const float*)d_in[1];
  float* out = (float*)d_out;
  const int T = in_sizes[0];
  const int nsteps = T - 1;
  unsigned* flags = (unsigned*)d_ws;   // (T-1+31)/32 words = 128 KB for T=2^20

  // Pass 1: serial h scan (single wave; inherently sequential critical path)
  h_scan_kernel<<<1, 32, 0, stream>>>(X, params, out, flags, T);

  // Pass 2: parallel WMMA-based J/H reconstruction over 16-step chunks
  const int nchunks = (nsteps + 15) / 16;
  jh_wmma_kernel<<<nchunks, 32, 0, stream>>>(X, params, out, flags, T);
}